// TensorRing_Based_4183298146465
// MI455X (gfx1250) — compile-verified
//
#include <hip/hip_runtime.h>

typedef __attribute__((ext_vector_type(2))) float v2f;
typedef __attribute__((ext_vector_type(4))) float v4f;
typedef __attribute__((ext_vector_type(8))) float v8f;

#define B_TOT 512
#define NF    8
#define DD    4
#define RR    32
#define UU    64

// kernel tensor strides (floats): [D=4, R=32, R=32, F=8, U=64]
//   d: 524288, i: 16384, j: 512, f: 64, u: 1
// LDS slice layout for fixed u: [f][i][j][d] -> idx = f*4096 + i*128 + j*4 + d
__device__ __forceinline__ float t_elem(const float* sk, v4f xf, int f, int i, int j) {
  const v4f kv = *(const v4f*)(sk + f * 4096 + i * 128 + j * 4);
  return fmaf(xf.x, kv.x, fmaf(xf.y, kv.y, fmaf(xf.z, kv.z, xf.w * kv.w)));
}

extern "C" __global__ __launch_bounds__(256)
void tensor_ring_wmma(const float* __restrict__ X, const float* __restrict__ Kg,
                      float* __restrict__ out) {
  extern __shared__ float smem[];
  const int u    = blockIdx.x;        // 0..63
  const int bg   = blockIdx.y;        // 0..15, covers b in [bg*32, bg*32+32)
  const int tid  = threadIdx.x;
  const int lane = tid & 31;
  const int wave = tid >> 5;          // 8 waves per block
  const int h    = lane >> 4;         // lane half (0/1)
  const int nl   = lane & 15;         // lane within half

  // ---- stage kernel slice K[:,:,:,:,u] into LDS as [f][i][j][d] (128 KB) ----
  for (int idx = tid; idx < NF * RR * RR * DD; idx += 256) {
    int d = idx & 3, j = (idx >> 2) & 31, i = (idx >> 7) & 31, f = idx >> 12;
    smem[idx] = Kg[d * 524288 + i * 16384 + j * 512 + f * 64 + u];
  }
  __syncthreads();

  // wave-private 32x32 transpose staging buffer (4 KB per wave)
  float* stage = smem + NF * RR * RR * DD + wave * (RR * RR);

  #pragma unroll 1
  for (int bi = wave; bi < 32; bi += 8) {
    const int b = bg * 32 + bi;

    // M accumulator: 2x2 tiles of 16x16 f32 in WMMA C/D layout.
    // Tile (a,bt), reg r: this lane holds element (m = 16a + r + 8h, n = 16bt + nl).
    v8f acc[2][2];
    {
      // M = T_0, built directly in C layout
      v4f x0 = *(const v4f*)(X + b * 32 + 0 * 4);
      #pragma unroll
      for (int a = 0; a < 2; ++a)
        #pragma unroll
        for (int bt = 0; bt < 2; ++bt)
          #pragma unroll
          for (int r = 0; r < 8; ++r)
            acc[a][bt][r] = t_elem(smem, x0, 0, 16 * a + r + 8 * h, 16 * bt + nl);
    }

    // ---- chain: M = M * T_f, f = 1..6 (last factor fused into the trace) ----
    #pragma unroll 1
    for (int f = 1; f < NF - 1; ++f) {
      // 1) spill M (C layout) to LDS row-major; same-wave LDS ops are in-order
      #pragma unroll
      for (int a = 0; a < 2; ++a)
        #pragma unroll
        for (int bt = 0; bt < 2; ++bt)
          #pragma unroll
          for (int r = 0; r < 8; ++r)
            stage[(16 * a + r + 8 * h) * 32 + 16 * bt + nl] = acc[a][bt][r];

      // 2) reload M as A-operands (16x4 f32 layout): for k-chunk kc, this lane
      //    needs (m = 16a + nl, k = 4kc + 2h {, +1}) -> contiguous pair -> b64 load
      v2f A[2][8];
      #pragma unroll
      for (int a = 0; a < 2; ++a)
        #pragma unroll
        for (int kc = 0; kc < 8; ++kc)
          A[a][kc] = *(const v2f*)(stage + (16 * a + nl) * 32 + 4 * kc + 2 * h);

      // 3) build T_f in B layout on the fly and accumulate with fp32 WMMA
      v4f xf = *(const v4f*)(X + b * 32 + f * 4);
      v8f nacc[2][2] = {};
      #pragma unroll
      for (int kc = 0; kc < 8; ++kc) {
        v2f Bv[2];
        #pragma unroll
        for (int bt = 0; bt < 2; ++bt) {
          // B 4x16 layout: VGPR0 holds k = 4kc+2h, VGPR1 holds k = 4kc+2h+1, n = lane
          int k0 = 4 * kc + 2 * h;
          int n  = 16 * bt + nl;
          Bv[bt].x = t_elem(smem, xf, f, k0, n);
          Bv[bt].y = t_elem(smem, xf, f, k0 + 1, n);
        }
        #pragma unroll
        for (int a = 0; a < 2; ++a)
          #pragma unroll
          for (int bt = 0; bt < 2; ++bt)
            nacc[a][bt] = __builtin_amdgcn_wmma_f32_16x16x4_f32(
                false, A[a][kc], false, Bv[bt], (short)0, nacc[a][bt], false, false);
      }
      #pragma unroll
      for (int a = 0; a < 2; ++a)
        #pragma unroll
        for (int bt = 0; bt < 2; ++bt)
          acc[a][bt] = nacc[a][bt];
    }

    // ---- fused last step + trace:
    // out[b,u] = trace(M * T7) = sum_{m,n} M[m,n] * T7[n,m].
    // Each lane dots its 32 resident C-layout elements against T7^T in place. ----
    float dsum = 0.f;
    {
      v4f x7 = *(const v4f*)(X + b * 32 + (NF - 1) * 4);
      #pragma unroll
      for (int a = 0; a < 2; ++a)
        #pragma unroll
        for (int bt = 0; bt < 2; ++bt)
          #pragma unroll
          for (int r = 0; r < 8; ++r) {
            int m = 16 * a + r + 8 * h;
            int n = 16 * bt + nl;
            dsum = fmaf(acc[a][bt][r], t_elem(smem, x7, NF - 1, n, m), dsum);
          }
    }

    // wave32 reduction
    #pragma unroll
    for (int off = 16; off > 0; off >>= 1)
      dsum += __shfl_xor(dsum, off, 32);
    if (lane == 0) out[b * 64 + u] = dsum;
  }
}

extern "C" void kernel_launch(void* const* d_in, const int* in_sizes, int n_in,
                              void* d_out, int out_size, void* d_ws, size_t ws_size,
                              hipStream_t stream) {
  (void)in_sizes; (void)n_in; (void)d_ws; (void)ws_size; (void)out_size;
  const float* X  = (const float*)d_in[0];   // [512, 8, 4] fp32
  const float* Kg = (const float*)d_in[1];   // [4, 32, 32, 8, 64] fp32
  float* out      = (float*)d_out;           // [512, 64] fp32

  const size_t shmem = (size_t)(NF * RR * RR * DD + 8 * RR * RR) * sizeof(float); // 160 KB
  (void)hipFuncSetAttribute((const void*)tensor_ring_wmma,
                            hipFuncAttributeMaxDynamicSharedMemorySize, (int)shmem);
  dim3 grid(UU, 16);   // 64 u-slices x 16 b-groups
  dim3 block(256);     // 8 wave32s
  hipLaunchKernelGGL(tensor_ring_wmma, grid, block, shmem, stream, X, Kg, out);
}